// Model_10977936408652
// MI455X (gfx1250) — compile-verified
//
#include <hip/hip_runtime.h>

#define NT 256
#define NE 512
#define NB 32
#define DIM 64
#define ECH 256   // e-chunk staged in LDS
#define TCH 16    // t values per block
#define EPAD 68   // padded row stride (dwords) to avoid LDS bank conflicts on b128 reads

typedef __attribute__((ext_vector_type(2))) float v2f;
typedef __attribute__((ext_vector_type(8))) float v8f;

// ---------------- Phase 1: P(R,64) = X(R,64) @ W(64,64), fp32 WMMA ----------------
// One wave per 16x16 output tile; K marched 4 at a time via V_WMMA_F32_16X16X4_F32.
__global__ __launch_bounds__(32) void proj_gemm_wmma(const float* __restrict__ X,
                                                     const float* __restrict__ W,
                                                     float* __restrict__ P) {
  const int lane = threadIdx.x;           // 0..31, full wave, EXEC all ones
  const int tcol = blockIdx.x & 3;        // 4 column tiles (64/16)
  const int trow = blockIdx.x >> 2;
  const int half = lane >> 4;             // 0: lanes 0-15, 1: lanes 16-31
  const int l15  = lane & 15;
  const int m    = trow * 16 + l15;       // A row for this lane
  const int n    = tcol * 16 + l15;       // B/C column for this lane
  const int koff = half * 2;              // lanes 16-31 hold K+2,K+3

  v8f acc = {};
  #pragma unroll
  for (int k0 = 0; k0 < DIM; k0 += 4) {
    v2f a, bf;
    // A 16x4 fp32: VGPR0 = K (lo lanes) / K+2 (hi lanes); VGPR1 = K+1 / K+3
    a.x = X[(size_t)m * DIM + k0 + koff];
    a.y = X[(size_t)m * DIM + k0 + koff + 1];
    // B 4x16 fp32: rows striped across lanes within a VGPR, split across lane halves
    bf.x = W[(size_t)(k0 + koff) * DIM + n];
    bf.y = W[(size_t)(k0 + koff + 1) * DIM + n];
    acc = __builtin_amdgcn_wmma_f32_16x16x4_f32(false, a, false, bf,
                                                (short)0, acc, false, false);
  }
  // C/D 16x16 fp32: VGPR v -> row (v + half*8), col n
  const int mbase = trow * 16 + half * 8;
  #pragma unroll
  for (int v = 0; v < 8; ++v)
    P[(size_t)(mbase + v) * DIM + n] = acc[v];
}

// ---------------- fast tanh: hardware V_TANH_F32 if available ----------------
__device__ __forceinline__ float fast_tanh(float x) {
#if __has_builtin(__builtin_amdgcn_tanhf)
  return __builtin_amdgcn_tanhf(x);
#else
  // tanh(x) = 1 - 2/(exp(2x)+1); exp via v_exp_f32, divide via v_rcp_f32
  float e = __builtin_amdgcn_exp2f(x * 2.885390081777926774f); // 2*log2(e)
  return 1.0f - 2.0f * __builtin_amdgcn_rcpf(e + 1.0f);
#endif
}

// ---------------- Phase 2: scores + masked mean, tanh-bound ----------------
__global__ __launch_bounds__(256) void score_kernel(
    const float* __restrict__ t_proj, const float* __restrict__ e_proj,
    const float* __restrict__ mask, const float* __restrict__ b1,
    const float* __restrict__ W2, const float* __restrict__ b2p,
    float* __restrict__ out_scores, float* __restrict__ out_sl) {
  __shared__ float e_lds[ECH * EPAD];     // e_proj[b] chunk, [e][d] padded
  __shared__ float tpb[TCH * EPAD];       // t_proj[t,b,:] + b1
  __shared__ float w2s[DIM];
  __shared__ float red[17 * 256];         // 16 t-partials + mask-count

  const int tid = threadIdx.x;
  const int t0  = blockIdx.x * TCH;
  const int b   = blockIdx.y;
  const float b2 = b2p[0];

  // stage t_proj chunk (+b1) and W2
  for (int idx = tid; idx < TCH * DIM; idx += 256) {
    int tl = idx >> 6, d = idx & 63;
    tpb[tl * EPAD + d] = t_proj[((size_t)(t0 + tl) * NB + b) * DIM + d] + b1[d];
  }
  if (tid < DIM) w2s[tid] = W2[tid];

  float acc[TCH];
  #pragma unroll
  for (int t = 0; t < TCH; ++t) acc[t] = 0.f;
  float cntp = 0.f;

  for (int e0 = 0; e0 < NE; e0 += ECH) {
    __syncthreads();
    // stage e_proj[e0..e0+255, b, :] into LDS (float4 global loads, coalesced)
    for (int v = tid; v < ECH * (DIM / 4); v += 256) {
      int el = v >> 4, d4 = v & 15;
      float4 src = *(const float4*)&e_proj[((size_t)(e0 + el) * NB + b) * DIM + d4 * 4];
      *(float4*)&e_lds[el * EPAD + d4 * 4] = src;
    }
    __syncthreads();

    const int e = e0 + tid;                 // one e per thread
    const float mk = mask[(size_t)b * NE + e];
    const float msel = (mk != 0.f) ? 1.f : 0.f;
    cntp += msel;

    #pragma unroll 1
    for (int t = 0; t < TCH; ++t) {
      float s = 0.f;
      #pragma unroll
      for (int d4 = 0; d4 < DIM / 4; ++d4) {
        float4 tp = *(const float4*)&tpb[t * EPAD + d4 * 4];     // broadcast
        float4 ep = *(const float4*)&e_lds[tid * EPAD + d4 * 4]; // conflict-free
        float4 w  = *(const float4*)&w2s[d4 * 4];                // broadcast
        s += w.x * fast_tanh(tp.x + ep.x);
        s += w.y * fast_tanh(tp.y + ep.y);
        s += w.z * fast_tanh(tp.z + ep.z);
        s += w.w * fast_tanh(tp.w + ep.w);
      }
      s += b2;
      out_sl[((size_t)b * NT + (t0 + t)) * NE + e] = s;  // coalesced over e
      acc[t] += msel * s;
    }
  }

  // block reduction for masked mean
  __syncthreads();
  #pragma unroll
  for (int t = 0; t < TCH; ++t) red[t * 256 + tid] = acc[t];
  red[16 * 256 + tid] = cntp;
  __syncthreads();
  for (int srd = 128; srd > 0; srd >>= 1) {
    if (tid < srd) {
      #pragma unroll
      for (int r = 0; r < 17; ++r) red[r * 256 + tid] += red[r * 256 + tid + srd];
    }
    __syncthreads();
  }
  if (tid < TCH)
    out_scores[(size_t)b * NT + t0 + tid] = red[tid * 256] / red[16 * 256];
}

extern "C" void kernel_launch(void* const* d_in, const int* in_sizes, int n_in,
                              void* d_out, int out_size, void* d_ws, size_t ws_size,
                              hipStream_t stream) {
  const float* targets    = (const float*)d_in[0];  // (256,32,64)
  const float* embeddings = (const float*)d_in[1];  // (512,32,64)
  const float* mask       = (const float*)d_in[2];  // (32,512)
  const float* W1         = (const float*)d_in[3];  // (128,64)
  const float* b1         = (const float*)d_in[4];  // (64,)
  const float* W2         = (const float*)d_in[5];  // (64,)
  const float* b2         = (const float*)d_in[6];  // scalar

  float* out_scores = (float*)d_out;                // (32,256)
  float* out_sl     = out_scores + (size_t)NB * NT; // (32,256,512)

  float* t_proj = (float*)d_ws;                         // 8192x64
  float* e_proj = t_proj + (size_t)NT * NB * DIM;       // 16384x64

  // t_proj = targets @ W1[:64];  e_proj = embeddings @ W1[64:]
  proj_gemm_wmma<<<dim3((NT * NB / 16) * 4), 32, 0, stream>>>(targets, W1, t_proj);
  proj_gemm_wmma<<<dim3((NE * NB / 16) * 4), 32, 0, stream>>>(embeddings, W1 + DIM * DIM, e_proj);

  score_kernel<<<dim3(NT / TCH, NB), 256, 0, stream>>>(
      t_proj, e_proj, mask, b1, W2, b2, out_scores, out_sl);
}